// RCModel_46488726011921
// MI455X (gfx1250) — compile-verified
//
#include <hip/hip_runtime.h>
#include <hip/hip_bf16.h>
#include <math.h>

// ---------------------------------------------------------------------------
// Problem constants (match reference)
// ---------------------------------------------------------------------------
#define NN 128         // 2 + 126 states
#define MM 127         // 1 + 126 inputs (b_T column + 126 rooms)
#define NR 126         // N_ROOMS
#define KTAB 1700      // Tout table length
#define HSTEP 30.0f    // dt (constant in t_eval)
#define TOUT_DT 3600.0f

// ext vector types for WMMA
typedef __attribute__((ext_vector_type(2))) float v2f;
typedef __attribute__((ext_vector_type(8))) float v8f;

// ws layout (in floats)
#define WS_PHIT   0            // 128*128  PhiT[j*128+i] = Phi[i][j]
#define WS_A2     16384        // 128*128
#define WS_A3     32768        // 128*128
#define WS_A4     49152        // 128*128
#define WS_BT     65536        // 128
#define WS_BQ     65664        // 128
#define WS_VECS   65792        // 5*128  (v1,v2,v3,v4,vq)

// ---------------------------------------------------------------------------
// 1) b_T = B[:,0];  b_q = B[:,1:] @ ((-cool*action + gain) * areas)
//    cool = sigmoid(loads[0])*500 ; gain = sigmoid(loads[1])*500
//    loads is (2, 126); Q_watts has 126 elements.
// ---------------------------------------------------------------------------
__global__ void prep_bvec(const float* __restrict__ B,
                          const float* __restrict__ loads,
                          const float* __restrict__ areas,
                          const int*   __restrict__ action,
                          float* __restrict__ ws) {
  __shared__ float sQ[NR];
  int i = threadIdx.x;          // 0..127
  float act = (float)action[0];
  if (i < NR) {
    float c = 500.0f / (1.0f + __expf(-loads[i]));        // loads[0][i]
    float g = 500.0f / (1.0f + __expf(-loads[NR + i]));   // loads[1][i]
    sQ[i] = (-c * act + g) * areas[i];
  }
  __syncthreads();
  // row i of B
  float bq = 0.0f;
  const float* Brow = B + (size_t)i * MM;
  for (int j = 0; j < NR; ++j) bq += Brow[1 + j] * sQ[j];
  ws[WS_BT + i] = Brow[0];
  ws[WS_BQ + i] = bq;
}

// ---------------------------------------------------------------------------
// 2) 128x128x128 GEMM  C = Aop @ Bop  using v_wmma_f32_16x16x4_f32.
//    Grid (8,8), 32 threads (one wave) per block, one 16x16 C tile per block.
// ---------------------------------------------------------------------------
__global__ __launch_bounds__(32)
void gemm128_wmma(const float* __restrict__ Aop,
                  const float* __restrict__ Bop,
                  float* __restrict__ C) {
  const int lane = threadIdx.x;        // 0..31
  const int m    = lane & 15;
  const int h    = lane >> 4;          // 0 or 1
  const int c0   = blockIdx.x * 16;    // N tile
  const int r0   = blockIdx.y * 16;    // M tile

  v8f acc = {};
#if __has_builtin(__builtin_amdgcn_wmma_f32_16x16x4_f32)
  for (int k0 = 0; k0 < NN; k0 += 4) {
    v2f a, b;
    // A 16x4 tile: lane holds A[r0+m][k0 + 2h + v]
    a[0] = Aop[(size_t)(r0 + m) * NN + k0 + 2 * h + 0];
    a[1] = Aop[(size_t)(r0 + m) * NN + k0 + 2 * h + 1];
    // B 4x16 tile: lane holds B[k0 + 2h + v][c0+m]
    b[0] = Bop[(size_t)(k0 + 2 * h + 0) * NN + c0 + m];
    b[1] = Bop[(size_t)(k0 + 2 * h + 1) * NN + c0 + m];
    acc = __builtin_amdgcn_wmma_f32_16x16x4_f32(
        /*neg_a=*/false, a, /*neg_b=*/false, b,
        /*c_mod=*/(short)0, acc, /*reuse_a=*/false, /*reuse_b=*/false);
  }
#else
  for (int k = 0; k < NN; ++k) {
    float bv = Bop[(size_t)k * NN + c0 + m];
    #pragma unroll
    for (int r = 0; r < 8; ++r)
      acc[r] += Aop[(size_t)(r0 + r + 8 * h) * NN + k] * bv;
  }
#endif
  // C/D layout: vgpr r -> row r + 8h, col = m
  #pragma unroll
  for (int r = 0; r < 8; ++r)
    C[(size_t)(r0 + r + 8 * h) * NN + c0 + m] = acc[r];
}

// ---------------------------------------------------------------------------
// 3) PhiT[j*128+i] = I + h*A + h^2/2*A2 + h^3/6*A3 + h^4/24*A4   (transposed)
// ---------------------------------------------------------------------------
__global__ void build_phiT(const float* __restrict__ A, float* __restrict__ ws) {
  int idx = blockIdx.x * blockDim.x + threadIdx.x;   // 0..16383
  if (idx >= NN * NN) return;
  int i = idx >> 7;           // row
  int j = idx & 127;          // col
  float v = (i == j ? 1.0f : 0.0f)
          + 30.0f    * A[idx]
          + 450.0f   * ws[WS_A2 + idx]
          + 4500.0f  * ws[WS_A3 + idx]
          + 33750.0f * ws[WS_A4 + idx];
  ws[WS_PHIT + (size_t)j * NN + i] = v;
}

// ---------------------------------------------------------------------------
// 4) Forcing vectors (exact 3/8-rule algebra for affine f):
//    g = (h/8)[(I + hA + h^2/3 A^2 + h^3/3 A^3) u1
//              + (3I + 2hA + h^2 A^2) u2 + (3I + hA) u3 + u4]
//    with u_i = b_T*Tout_i + b_q  =>
//    v1 = 3.75*(bT + 30 w1 + 300 w2 + 9000 w3)
//    v2 = 3.75*(3 bT + 60 w1 + 900 w2)
//    v3 = 3.75*(3 bT + 30 w1)
//    v4 = 3.75* bT
//    vq = 3.75*(8 bq + 120 q1 + 1200 q2 + 9000 q3)
//    where w_k = A^k bT, q_k = A^k bq.
// ---------------------------------------------------------------------------
__global__ __launch_bounds__(NN)
void vec_chain(const float* __restrict__ A, float* __restrict__ ws) {
  __shared__ float sin_[NN];
  int i = threadIdx.x;
  const float* Arow = A + (size_t)i * NN;

  auto matvec = [&](void) {
    float y = 0.0f;
    #pragma unroll 8
    for (int j = 0; j < NN; ++j) y += Arow[j] * sin_[j];
    return y;
  };

  float bT = ws[WS_BT + i];
  sin_[i] = bT; __syncthreads();
  float w1 = matvec(); __syncthreads(); sin_[i] = w1; __syncthreads();
  float w2 = matvec(); __syncthreads(); sin_[i] = w2; __syncthreads();
  float w3 = matvec(); __syncthreads();

  float bq = ws[WS_BQ + i];
  sin_[i] = bq; __syncthreads();
  float q1 = matvec(); __syncthreads(); sin_[i] = q1; __syncthreads();
  float q2 = matvec(); __syncthreads(); sin_[i] = q2; __syncthreads();
  float q3 = matvec();

  ws[WS_VECS + 0 * NN + i] = 3.75f * (bT + 30.0f * w1 + 300.0f * w2 + 9000.0f * w3);
  ws[WS_VECS + 1 * NN + i] = 3.75f * (3.0f * bT + 60.0f * w1 + 900.0f * w2);
  ws[WS_VECS + 2 * NN + i] = 3.75f * (3.0f * bT + 30.0f * w1);
  ws[WS_VECS + 3 * NN + i] = 3.75f * bT;
  ws[WS_VECS + 4 * NN + i] = 3.75f * (8.0f * bq + 120.0f * q1 + 1200.0f * q2 + 9000.0f * q3);
}

// ---------------------------------------------------------------------------
// 5) Sequential integrator. One workgroup, 128 threads (4 waves, wave32,
//    one wave per SIMD). Thread i keeps Phi row i (128 f32) in VGPRs; x is
//    double-buffered in LDS. Four independent accumulators break the
//    dependent-FMA latency chain (latency-bound: 200k serial steps).
// ---------------------------------------------------------------------------
__device__ __forceinline__ float interp_tout(const float* t, float tabs) {
  float pos = tabs * (1.0f / TOUT_DT);
  int idx = (int)floorf(pos);
  idx = idx < 0 ? 0 : (idx > KTAB - 2 ? KTAB - 2 : idx);
  float w = pos - (float)idx;
  return t[idx] * (1.0f - w) + t[idx + 1] * w;
}

__global__ __launch_bounds__(NN, 1)
void integrate(const float* __restrict__ ws,
               const float* __restrict__ Tout_vals,
               const float* __restrict__ t_eval,
               const float* __restrict__ iv,
               float* __restrict__ out, int T) {
  __shared__ __align__(16) float xb[2][NN];
  __shared__ float tout[KTAB];
  const int i = threadIdx.x;

  for (int k = i; k < KTAB; k += NN) tout[k] = Tout_vals[k];

  // Phi row i into registers (coalesced: PhiT[j*128 + i])
  float phi[NN];
  #pragma unroll
  for (int j = 0; j < NN; ++j) phi[j] = ws[WS_PHIT + (size_t)j * NN + i];

  const float v1 = ws[WS_VECS + 0 * NN + i];
  const float v2 = ws[WS_VECS + 1 * NN + i];
  const float v3 = ws[WS_VECS + 2 * NN + i];
  const float v4 = ws[WS_VECS + 3 * NN + i];
  const float vq = ws[WS_VECS + 4 * NN + i];
  const float t0 = t_eval[0];

  float x = iv[i];
  out[i] = x;
  xb[0][i] = x;
  __syncthreads();

  int cur = 0;
  for (int n = 0; n < T - 1; ++n) {
    float tabs = t0 + HSTEP * (float)n;          // == t_eval[n] exactly
    float f0 = vq + v1 * interp_tout(tout, tabs);
    float f1 = v2 * interp_tout(tout, tabs + 10.0f);
    float f2 = v3 * interp_tout(tout, tabs + 20.0f);
    float f3 = v4 * interp_tout(tout, tabs + 30.0f);
    const float* xs = xb[cur];
    // 4 independent accumulator chains -> ~1 FMA/cycle issue per wave
    float a0 = f0, a1 = f1, a2 = f2, a3 = f3;
    #pragma unroll
    for (int j = 0; j < NN; j += 4) {
      float4 xv = *(const float4*)(xs + j);      // LDS broadcast read
      a0 += phi[j + 0] * xv.x;
      a1 += phi[j + 1] * xv.y;
      a2 += phi[j + 2] * xv.z;
      a3 += phi[j + 3] * xv.w;
    }
    float acc = (a0 + a1) + (a2 + a3);
    out[(size_t)(n + 1) * NN + i] = acc;
    xb[cur ^ 1][i] = acc;
    __syncthreads();
    cur ^= 1;
  }
}

// ---------------------------------------------------------------------------
// launcher
// ---------------------------------------------------------------------------
extern "C" void kernel_launch(void* const* d_in, const int* in_sizes, int n_in,
                              void* d_out, int out_size, void* d_ws, size_t ws_size,
                              hipStream_t stream) {
  const float* A        = (const float*)d_in[0];   // 128x128
  const float* B        = (const float*)d_in[1];   // 128x127
  const float* loads    = (const float*)d_in[2];   // 2x126
  const float* areas    = (const float*)d_in[3];   // 126
  const float* ToutVals = (const float*)d_in[4];   // 1700
  const float* t_eval   = (const float*)d_in[5];   // 200000
  const float* iv       = (const float*)d_in[6];   // 128
  const int*   action   = (const int*)d_in[7];     // 1
  float* out = (float*)d_out;
  float* ws  = (float*)d_ws;
  const int T = out_size / NN;                     // 200000

  prep_bvec<<<1, NN, 0, stream>>>(B, loads, areas, action, ws);

  dim3 gtile(8, 8);
  gemm128_wmma<<<gtile, 32, 0, stream>>>(A, A,            ws + WS_A2);  // A^2
  gemm128_wmma<<<gtile, 32, 0, stream>>>(A, ws + WS_A2,   ws + WS_A3);  // A^3
  gemm128_wmma<<<gtile, 32, 0, stream>>>(A, ws + WS_A3,   ws + WS_A4);  // A^4

  build_phiT<<<(NN * NN + 255) / 256, 256, 0, stream>>>(A, ws);
  vec_chain<<<1, NN, 0, stream>>>(A, ws);

  integrate<<<1, NN, 0, stream>>>(ws, ToutVals, t_eval, iv, out, T);
}